// HdcLevelEncoder_13159779795489
// MI455X (gfx1250) — compile-verified
//
#include <hip/hip_runtime.h>
#include <stdint.h>

#define D_DIM   100000
#define N_SAMP  128
#define NLEVELS 100
#define NTS     128
#define TILE_D  128
#define BLOCK   128
#define FBIG    91
#define W_ELT   (TILE_D * FBIG)   /* 11648 floats of W_big tile in LDS */

typedef float v2f __attribute__((ext_vector_type(2)));
typedef float v8f __attribute__((ext_vector_type(8)));
typedef int   v4i __attribute__((ext_vector_type(4)));

#if __has_builtin(__builtin_amdgcn_wmma_f32_16x16x4_f32)
#define HAS_WMMA4 1
#else
#define HAS_WMMA4 0
#endif

#if __has_builtin(__builtin_amdgcn_global_load_async_to_lds_b128)
#define HAS_ASYNC 1
#else
#define HAS_ASYNC 0
#endif

__device__ __forceinline__ int quant_idx(float x, float lo, float hi, int num) {
  float xc = fminf(fmaxf(x, lo), hi);
  float fi = rintf((xc - lo) / (hi - lo) * (float)(num - 1));
  int i = (int)fi;
  i = i < 0 ? 0 : i;
  i = i > num - 1 ? num - 1 : i;
  return i;
}

__global__ __launch_bounds__(BLOCK) void hdc_fused(
    const float* __restrict__ inp,  const float* __restrict__ feat,
    const float* __restrict__ lvx,  const float* __restrict__ lvy,
    const float* __restrict__ lvz,  const float* __restrict__ lvt,
    const float* __restrict__ Wb,   const float* __restrict__ bb,
    const float* __restrict__ Ws,   const float* __restrict__ bs,
    float* __restrict__ out)
{
  __shared__ __align__(16) float sW[W_ELT + 8];
  __shared__ float sPB[TILE_D];
  __shared__ float sFeat[600];
  __shared__ int   sIx[N_SAMP], sIy[N_SAMP], sIz[N_SAMP], sIt[N_SAMP];

  const int tid  = threadIdx.x;
  const int lane = tid & 31;
  const int d0   = blockIdx.x * TILE_D;
  const int d    = d0 + tid;
  const bool dok = (d < D_DIM);
  const int dcount = (D_DIM - d0 < TILE_D) ? (D_DIM - d0) : TILE_D;

  // ---- per-block setup: sample indices + feat into LDS ----
  if (tid < N_SAMP) {
    float t = inp[tid * 4 + 0];
    float x = inp[tid * 4 + 1];
    float y = inp[tid * 4 + 2];
    float z = inp[tid * 4 + 3];
    sIt[tid] = quant_idx(t, 0.f, (float)NTS, NTS);
    sIx[tid] = quant_idx(x, -5.f, 5.f, NLEVELS);
    sIy[tid] = quant_idx(y, -5.f, 5.f, NLEVELS);
    sIz[tid] = quant_idx(z, -5.f, 5.f, NLEVELS);
  }
  for (int i = tid; i < 600; i += BLOCK) sFeat[i] = feat[i];
  if (tid < 8) sW[W_ELT + tid] = 0.f;   // pad beyond tile, keep finite
  __syncthreads();

  // ---- phase A: sign of prod_n (x+y+z)*t (values are +-1/+-3, never 0) ----
  uint32_t sbits = 0u;
  if (dok) {
    for (int n = 0; n < N_SAMP; ++n) {
      float vx = lvx[(size_t)sIx[n] * D_DIM + d];
      float vy = lvy[(size_t)sIy[n] * D_DIM + d];
      float vz = lvz[(size_t)sIz[n] * D_DIM + d];
      float vt = lvt[(size_t)sIt[n] * D_DIM + d];
      float v = (vx + vy + vz) * vt;
      sbits ^= __float_as_uint(v);      // accumulate sign bit
    }
  }
  const float hsign = (sbits & 0x80000000u) ? -1.f : 1.f;

  // ---- phase B: pb[k,d] = sum_f feat_big[k,f]*W_big[k,d,f] via WMMA ----
  const int mbase = (tid >> 5) << 5;    // each wave owns 32 rows (two M-tiles)
  const int mrow  = lane & 15;          // A: lanes 0-15 => M=0..15 (both halves)
  const int koff  = (lane >> 4) << 1;   // A/B: high lane half carries K+2
  float prodBig = 1.f;

  for (int k = 0; k < 6; ++k) {
    const float* gsrc = Wb + ((size_t)k * D_DIM + (size_t)d0) * FBIG;
    const int nchunk = (dcount * FBIG) >> 2;   // 16B chunks (dcount % 4 == 0)
#if HAS_ASYNC
    {
      typedef __attribute__((address_space(1))) v4i g1_v4i;
      typedef __attribute__((address_space(3))) v4i l3_v4i;
      g1_v4i* gp = (g1_v4i*)gsrc;   // cast away const; builtin takes non-const
      l3_v4i* lp = (l3_v4i*)sW;
      for (int c = tid; c < nchunk; c += BLOCK)
        __builtin_amdgcn_global_load_async_to_lds_b128(gp + c, lp + c, 0, 0);
#if __has_builtin(__builtin_amdgcn_s_wait_asynccnt)
      __builtin_amdgcn_s_wait_asynccnt(0);
#else
      asm volatile("s_wait_asynccnt 0" ::: "memory");
#endif
    }
#else
    {
      const float4* g4 = (const float4*)gsrc;
      float4* s4 = (float4*)sW;
      for (int c = tid; c < nchunk; c += BLOCK) s4[c] = g4[c];
    }
#endif
    __syncthreads();

#if HAS_WMMA4
    v8f c0 = {0.f, 0.f, 0.f, 0.f, 0.f, 0.f, 0.f, 0.f};
    v8f c1 = c0;
    const float* fk = &sFeat[k * FBIG];
    for (int f = 0; f < FBIG; f += 4) {
      const int fa = f + koff;                 // <= 90 always
      const bool okB = (fa + 1 < FBIG);        // K-tail zero pad
      v2f bvec;                                // B broadcast across N
      bvec.x = fk[fa];
      bvec.y = okB ? fk[fa + 1] : 0.f;
      const int r0 = (mbase + mrow) * FBIG + fa;
      const int r1 = (mbase + 16 + mrow) * FBIG + fa;
      v2f a0, a1;
      a0.x = sW[r0];
      a0.y = okB ? sW[r0 + 1] : 0.f;
      a1.x = sW[r1];
      a1.y = okB ? sW[r1 + 1] : 0.f;
      c0 = __builtin_amdgcn_wmma_f32_16x16x4_f32(false, a0, false, bvec,
                                                 (short)0, c0, false, false);
      c1 = __builtin_amdgcn_wmma_f32_16x16x4_f32(false, a1, false, bvec,
                                                 (short)0, c1, false, false);
    }
    // D layout: lane L, VGPR v -> M = v + 8*(L>=16); all N columns identical.
    if ((lane & 15) == 0) {
      const int half = (lane >> 4) << 3;       // 0 or 8
#pragma unroll
      for (int v = 0; v < 8; ++v) {
        sPB[mbase + half + v]      = c0[v];
        sPB[mbase + 16 + half + v] = c1[v];
      }
    }
#else
    {
      float acc = 0.f;
      const float* fk = &sFeat[k * FBIG];
      const float* wr = &sW[tid * FBIG];
      for (int f = 0; f < FBIG; ++f) acc = fmaf(wr[f], fk[f], acc);
      sPB[tid] = acc;
    }
#endif
    __syncthreads();

    if (dok) {
      float pbv = sPB[tid];
      float bbv = bb[(size_t)k * D_DIM + d];
      prodBig *= __cosf(pbv + bbv) * __sinf(pbv);
    }
  }

  // ---- phase C: the 9 needed small-k features + combination tree ----
  if (dok) {
    const float* fe = &sFeat[546];
    auto smallf = [&](int kk) -> float {
      const float* w = Ws + ((size_t)kk * D_DIM + (size_t)d) * 3;
      float ps = fmaf(w[0], fe[kk * 3 + 0],
                 fmaf(w[1], fe[kk * 3 + 1], w[2] * fe[kk * 3 + 2]));
      float bv = bs[(size_t)kk * D_DIM + d];
      return __cosf(ps + bv) * __sinf(ps);
    };
    float f6  = smallf(0);
    float f9  = smallf(3);
    float f10 = smallf(4);
    float f11 = smallf(5);
    float f12 = smallf(6);
    float f17 = smallf(11);
    float f18 = smallf(12);
    float f21 = smallf(15);
    float f23 = smallf(17);
    float t1 = (f6 + f21 + f23) * (f9 + f10) * f11 * f12 * f17 * f18;
    float t2 = (f6 + f10 + f11 + f12) * prodBig;
    float val = hsign * (t1 + t2);
    out[d] = (val > 0.f) ? 1.f : -1.f;
  }
}

extern "C" void kernel_launch(void* const* d_in, const int* in_sizes, int n_in,
                              void* d_out, int out_size, void* d_ws, size_t ws_size,
                              hipStream_t stream) {
  const float* inp  = (const float*)d_in[0];
  const float* feat = (const float*)d_in[1];
  const float* lvx  = (const float*)d_in[2];
  const float* lvy  = (const float*)d_in[3];
  const float* lvz  = (const float*)d_in[4];
  const float* lvt  = (const float*)d_in[5];
  const float* Wb   = (const float*)d_in[6];
  const float* bb   = (const float*)d_in[7];
  const float* Ws   = (const float*)d_in[8];
  const float* bs   = (const float*)d_in[9];
  float* out = (float*)d_out;
  (void)in_sizes; (void)n_in; (void)out_size; (void)d_ws; (void)ws_size;

  const int grid = (D_DIM + TILE_D - 1) / TILE_D;   // 782 blocks
  hipLaunchKernelGGL(hdc_fused, dim3(grid), dim3(BLOCK), 0, stream,
                     inp, feat, lvx, lvy, lvz, lvt, Wb, bb, Ws, bs, out);
}